// integralDecoder_28260884807699
// MI455X (gfx1250) — compile-verified
//
#include <hip/hip_runtime.h>
#include <hip/hip_bf16.h>

typedef __attribute__((ext_vector_type(2))) float v2f;
typedef __attribute__((ext_vector_type(4))) float v4f;
typedef __attribute__((ext_vector_type(8))) float v8f;

#define TWO_N 8192
#define N_POP 4096
#define SCALED_POP_MUT 0.0016384f   // 4 * N * U = 4 * 4096 * 1e-7
#define EPS_F 1e-6f
#define NWAVES 8
#define KSPAN (TWO_N / NWAVES)      // 1024 columns per wave

// ---------------------------------------------------------------------------
// Kernel 1: v[j] = final_prf(points[j]; s) * trapezoid_weight[j]
// ---------------------------------------------------------------------------
__global__ __launch_bounds__(256) void build_v_kernel(
    const float* __restrict__ s_ptr,
    const float* __restrict__ points,
    float* __restrict__ v)
{
    int j = blockIdx.x * blockDim.x + threadIdx.x;
    if (j >= TWO_N) return;

    float s      = *s_ptr;
    float scaled = 2.0f * (float)N_POP * s;
    float denom  = __expf(-scaled);
    float p      = points[j];

    float ratio  = (1.0f - __expf(-scaled * (1.0f - p))) / (1.0f - __expf(-scaled));
    float neg_c  = isinf(denom)   ? (1.0f + fmaxf(s, EPS_F)) : ratio;
    float pos_c  = (denom == 0.f) ? p                        : ratio;
    float finp   = (scaled < 0.f) ? neg_c : pos_c;

    // trapezoid weights for x = points
    float dr = (j + 1 < TWO_N) ? (points[j + 1] - points[j]) : 0.0f;
    float dl = (j > 0)         ? (points[j] - points[j - 1]) : 0.0f;

    v[j] = finp * 0.5f * (dl + dr);
}

// ---------------------------------------------------------------------------
// Kernel 2: out[i] = (i>0) ? log(SCALED_POP_MUT * dot(binom[i,:], v)) : 0
// One block = 16 rows. 8 waves, each wave owns a contiguous 1024-col K span.
// Reduction along K uses V_WMMA_F32_16X16X4_F32 with B broadcast across N.
// ---------------------------------------------------------------------------
__global__ __launch_bounds__(256) void matvec_wmma_kernel(
    const float* __restrict__ binom,
    const float* __restrict__ v,
    float* __restrict__ out)
{
    __shared__ float partial[NWAVES][16];

    const int tid     = threadIdx.x;
    const int lane    = tid & 31;
    const int wave    = tid >> 5;
    const int m       = lane & 15;        // row within 16-row tile
    const int halfSel = lane >> 4;        // 0: K slots {0,1}(+{2,3}), 1: K slots {4..7}
    const int rowBase = blockIdx.x * 16;

    const float* __restrict__ rowPtr =
        binom + (size_t)(rowBase + m) * (size_t)TWO_N;

    const int kBegin = wave * KSPAN;
    const int kEnd   = kBegin + KSPAN;

    v8f acc = {0.f, 0.f, 0.f, 0.f, 0.f, 0.f, 0.f, 0.f};

    for (int k0 = kBegin; k0 < kEnd; k0 += 8) {
        const int kc = k0 + 4 * halfSel;

        // speculative prefetch ~1KB ahead in this row (OOB is silently dropped)
        __builtin_prefetch((const void*)(rowPtr + kc + 256), 0, 1);

        v4f a = *(const v4f*)(rowPtr + kc);  // binom[row][kc .. kc+3]
        v4f b = *(const v4f*)(v + kc);       // v[kc .. kc+3], broadcast across N

        v2f a01 = {a[0], a[1]};
        v2f b01 = {b[0], b[1]};
        v2f a23 = {a[2], a[3]};
        v2f b23 = {b[2], b[3]};

        // D = A(16x4) x B(4x16) + C ; B columns all identical == v chunk,
        // so every accumulator element holds the row-partial dot product.
        acc = __builtin_amdgcn_wmma_f32_16x16x4_f32(
                  false, a01, false, b01, (short)0, acc, false, false);
        acc = __builtin_amdgcn_wmma_f32_16x16x4_f32(
                  false, a23, false, b23, (short)0, acc, false, false);
    }

    // C/D layout: VGPR r, lanes 0-15 -> M=r ; lanes 16-31 -> M=8+r.
    // All N columns identical, so lane 0 / lane 16 carry the answers.
    if (lane == 0) {
#pragma unroll
        for (int r = 0; r < 8; ++r) partial[wave][r] = acc[r];
    }
    if (lane == 16) {
#pragma unroll
        for (int r = 0; r < 8; ++r) partial[wave][8 + r] = acc[r];
    }
    __syncthreads();

    if (tid < 16) {
        float sum = 0.0f;
#pragma unroll
        for (int w = 0; w < NWAVES; ++w) sum += partial[w][tid];
        const int row = rowBase + tid;
        const float ints = SCALED_POP_MUT * sum;
        out[row] = (row > 0) ? logf(ints) : 0.0f;
    }
}

// ---------------------------------------------------------------------------
// Launch wrapper
// ---------------------------------------------------------------------------
extern "C" void kernel_launch(void* const* d_in, const int* in_sizes, int n_in,
                              void* d_out, int out_size, void* d_ws, size_t ws_size,
                              hipStream_t stream)
{
    const float* s      = (const float*)d_in[0];   // scalar selection coefficient
    const float* points = (const float*)d_in[1];   // [8192]
    const float* binom  = (const float*)d_in[2];   // [8192 x 8192]
    float*       out    = (float*)d_out;           // [8192]
    float*       v      = (float*)d_ws;            // [8192] scratch (32 KB)

    build_v_kernel<<<TWO_N / 256, 256, 0, stream>>>(s, points, v);
    matvec_wmma_kernel<<<TWO_N / 16, 256, 0, stream>>>(binom, v, out);
}